// GATConvBlock_3848290697222
// MI455X (gfx1250) — compile-verified
//
#include <hip/hip_runtime.h>
#include <math.h>

typedef float v2f __attribute__((ext_vector_type(2)));
typedef float v8f __attribute__((ext_vector_type(8)));

#define C_IN 128
#define NEG_SLOPE 0.2f
#define LN_EPS 1e-5f

// ---------------------------------------------------------------------------
// K0: init  out[n][d] = bias[d], mmax = -inf, denom = 0
// ---------------------------------------------------------------------------
__global__ void init_kernel(float* __restrict__ out, const float* __restrict__ bias,
                            float* __restrict__ mmax, float* __restrict__ denom,
                            long N) {
    long i = (long)blockIdx.x * blockDim.x + threadIdx.x;
    if (i < N * 32) out[i] = bias[i & 31];
    if (i < N * 4) { mmax[i] = -__builtin_inff(); denom[i] = 0.0f; }
}

// ---------------------------------------------------------------------------
// K1: fused LayerNorm + ReLU + dual GEMM (x_l = xh@W_l + b_l, x_r = xh@W_r + b_r)
// Block = 256 threads = 8 waves, handles 16 nodes (rows).
// LN: thread t -> row t/16, 8 channels; reductions via width-16 shuffles.
// GEMM: wave w owns 16-col tile w*16, K-loop of v_wmma_f32_16x16x4_f32 for both W.
// ---------------------------------------------------------------------------
__global__ __launch_bounds__(256) void ln_gemm_kernel(
    const float* __restrict__ x,
    const float* __restrict__ gamma, const float* __restrict__ beta,
    const float* __restrict__ Wl, const float* __restrict__ bl,
    const float* __restrict__ Wr, const float* __restrict__ br,
    float* __restrict__ xl, float* __restrict__ xr, long N)
{
    __shared__ float sXh[16][132];   // pad 132 -> A-frag column reads spread banks

    const int t   = threadIdx.x;
    const int row = t >> 4;
    const int sub = t & 15;
    const long node = (long)blockIdx.x * 16 + row;

    float v[8];
    if (node < N) {
        const float4* xp = (const float4*)(x + node * C_IN + sub * 8);
        float4 p0 = xp[0], p1 = xp[1];
        v[0]=p0.x; v[1]=p0.y; v[2]=p0.z; v[3]=p0.w;
        v[4]=p1.x; v[5]=p1.y; v[6]=p1.z; v[7]=p1.w;
    } else {
        #pragma unroll
        for (int j = 0; j < 8; ++j) v[j] = 0.0f;
    }

    // two-pass LayerNorm within each 16-lane half-wave (one row)
    float s = 0.0f;
    #pragma unroll
    for (int j = 0; j < 8; ++j) s += v[j];
    s += __shfl_xor(s, 1, 16); s += __shfl_xor(s, 2, 16);
    s += __shfl_xor(s, 4, 16); s += __shfl_xor(s, 8, 16);
    const float mu = s * (1.0f / 128.0f);

    float q = 0.0f;
    #pragma unroll
    for (int j = 0; j < 8; ++j) { float d = v[j] - mu; q += d * d; }
    q += __shfl_xor(q, 1, 16); q += __shfl_xor(q, 2, 16);
    q += __shfl_xor(q, 4, 16); q += __shfl_xor(q, 8, 16);
    const float rstd = rsqrtf(q * (1.0f / 128.0f) + LN_EPS);

    #pragma unroll
    for (int j = 0; j < 8; ++j) {
        const int c = sub * 8 + j;
        float y = (v[j] - mu) * rstd * gamma[c] + beta[c];
        sXh[row][c] = y > 0.0f ? y : 0.0f;   // ReLU
    }
    __syncthreads();

    // ---- WMMA phase ----
    const int lane = t & 31;
    const int wave = t >> 5;          // 0..7 -> col tile
    const int half = lane >> 4;       // 0: K=0,1 / 1: K=2,3
    const int lr   = lane & 15;       // A: row  B: col  (per ISA 16x16x4 layout)
    const int col  = wave * 16 + lr;

    v8f cl = {}; v8f cr = {};
    #pragma unroll 8
    for (int k = 0; k < 128; k += 4) {
        v2f a = *(const v2f*)&sXh[lr][k + 2 * half];     // A[lr][k+2h], A[lr][k+2h+1]
        const int rk = (k + 2 * half) * C_IN + col;
        v2f pb; pb.x = Wl[rk]; pb.y = Wl[rk + C_IN];     // B rows k+2h, k+2h+1 @ col
        cl = __builtin_amdgcn_wmma_f32_16x16x4_f32(false, a, false, pb, (short)0, cl, false, false);
        v2f qb; qb.x = Wr[rk]; qb.y = Wr[rk + C_IN];
        cr = __builtin_amdgcn_wmma_f32_16x16x4_f32(false, a, false, qb, (short)0, cr, false, false);
    }

    const float bll = bl[col], brr = br[col];
    const long base = (long)blockIdx.x * 16;
    #pragma unroll
    for (int vv = 0; vv < 8; ++vv) {
        const long r = base + vv + 8 * half;             // C/D: vgpr v -> row v + 8*half
        if (r < N) {
            xl[r * C_IN + col] = cl[vv] + bll;
            xr[r * C_IN + col] = cr[vv] + brr;
        }
    }
}

// ---------------------------------------------------------------------------
// float atomic max via signed/unsigned integer atomics (target init = -inf)
// ---------------------------------------------------------------------------
__device__ __forceinline__ void atomicMaxF(float* addr, float val) {
    if (val >= 0.0f) atomicMax((int*)addr, __float_as_int(val));
    else             atomicMin((unsigned int*)addr, __float_as_uint(val));
}

// ---------------------------------------------------------------------------
// K2: edge scores e = att . leaky_relu(x_l[src] + x_r[dst]) + segment max.
// One wave per edge: lane d loads float4 #d of each 128-float row (coalesced);
// head h = d/8; per-head dot reduced with width-8 xor shuffles.
// ---------------------------------------------------------------------------
__global__ __launch_bounds__(256) void edge_score_kernel(
    const int* __restrict__ srcI, const int* __restrict__ dstI,
    const float* __restrict__ xl, const float* __restrict__ xr,
    const float* __restrict__ att,
    float* __restrict__ escore, float* __restrict__ mmax,
    long E, long Etot)
{
    const int lane = threadIdx.x & 31;
    const long e = (long)blockIdx.x * 8 + (threadIdx.x >> 5);
    if (e >= Etot) return;
    int s_, d_;
    if (e < E) { s_ = srcI[e]; d_ = dstI[e]; }
    else       { s_ = d_ = (int)(e - E); }                 // self loop

    const float4 a4 = ((const float4*)att)[lane];
    const float4 l4 = ((const float4*)xl)[(long)s_ * 32 + lane];
    const float4 r4 = ((const float4*)xr)[(long)d_ * 32 + lane];

    float4 t4;
    t4.x = l4.x + r4.x; t4.y = l4.y + r4.y; t4.z = l4.z + r4.z; t4.w = l4.w + r4.w;
    t4.x = t4.x > 0.0f ? t4.x : NEG_SLOPE * t4.x;
    t4.y = t4.y > 0.0f ? t4.y : NEG_SLOPE * t4.y;
    t4.z = t4.z > 0.0f ? t4.z : NEG_SLOPE * t4.z;
    t4.w = t4.w > 0.0f ? t4.w : NEG_SLOPE * t4.w;

    float p = t4.x * a4.x + t4.y * a4.y + t4.z * a4.z + t4.w * a4.w;
    p += __shfl_xor(p, 1, 8);
    p += __shfl_xor(p, 2, 8);
    p += __shfl_xor(p, 4, 8);

    if ((lane & 7) == 0) {
        const int h = lane >> 3;
        escore[e * 4 + h] = p;
        atomicMaxF(&mmax[(long)d_ * 4 + h], p);
    }
}

// ---------------------------------------------------------------------------
// K3: w = exp(e - m[dst]); overwrite scores with w; denom[dst] += w
// ---------------------------------------------------------------------------
__global__ void softmax_denom_kernel(
    const int* __restrict__ dstI, float* __restrict__ escore,
    const float* __restrict__ mmax, float* __restrict__ denom,
    long E, long Etot)
{
    const long i = (long)blockIdx.x * blockDim.x + threadIdx.x;
    if (i >= Etot * 4) return;
    const long e = i >> 2;
    const int  h = (int)(i & 3);
    const int  d_ = (e < E) ? dstI[e] : (int)(e - E);
    const float w = __expf(escore[i] - mmax[(long)d_ * 4 + h]);
    escore[i] = w;
    atomicAdd(&denom[(long)d_ * 4 + h], w);
}

// ---------------------------------------------------------------------------
// K4: out[dst] += (1/H) * sum_h alpha_h * x_l[src][h]  (head-mean in-wave,
// only 32 atomics/edge into the final [N,32] output already holding bias)
// ---------------------------------------------------------------------------
__global__ __launch_bounds__(256) void aggregate_kernel(
    const int* __restrict__ srcI, const int* __restrict__ dstI,
    const float* __restrict__ xl, const float* __restrict__ escore,
    const float* __restrict__ denom, float* __restrict__ out,
    long E, long Etot)
{
    const int lane = threadIdx.x & 31;
    const long e = (long)blockIdx.x * 8 + (threadIdx.x >> 5);
    if (e >= Etot) return;
    int s_, d_;
    if (e < E) { s_ = srcI[e]; d_ = dstI[e]; }
    else       { s_ = d_ = (int)(e - E); }

    const int h = lane >> 3;
    const float w   = escore[e * 4 + h];
    const float den = denom[(long)d_ * 4 + h];
    const float alpha = 0.25f * w / (den + 1e-16f);        // 0.25 = head mean

    const float4 l4 = ((const float4*)xl)[(long)s_ * 32 + lane];
    float vx = alpha * l4.x, vy = alpha * l4.y, vz = alpha * l4.z, vw = alpha * l4.w;

    // sum the 4 heads: lanes {j, j+8, j+16, j+24} share the same in-head chunk j
    vx += __shfl_xor(vx, 8);  vy += __shfl_xor(vy, 8);
    vz += __shfl_xor(vz, 8);  vw += __shfl_xor(vw, 8);
    vx += __shfl_xor(vx, 16); vy += __shfl_xor(vy, 16);
    vz += __shfl_xor(vz, 16); vw += __shfl_xor(vw, 16);

    if (lane < 8) {
        float* o = out + (long)d_ * 32 + lane * 4;
        atomicAdd(o + 0, vx);
        atomicAdd(o + 1, vy);
        atomicAdd(o + 2, vz);
        atomicAdd(o + 3, vw);
    }
}

// ---------------------------------------------------------------------------
extern "C" void kernel_launch(void* const* d_in, const int* in_sizes, int n_in,
                              void* d_out, int out_size, void* d_ws, size_t ws_size,
                              hipStream_t stream) {
    const float* x     = (const float*)d_in[0];
    const int*   ei    = (const int*)d_in[1];
    const float* gamma = (const float*)d_in[2];
    const float* beta  = (const float*)d_in[3];
    const float* Wl    = (const float*)d_in[4];
    const float* bl    = (const float*)d_in[5];
    const float* Wr    = (const float*)d_in[6];
    const float* br    = (const float*)d_in[7];
    const float* att   = (const float*)d_in[8];
    const float* bias  = (const float*)d_in[9];
    float* out = (float*)d_out;

    const long N = in_sizes[0] / C_IN;
    const long E = in_sizes[1] / 2;
    const long Etot = E + N;
    const int* srcI = ei;
    const int* dstI = ei + E;

    float* ws     = (float*)d_ws;
    float* xl     = ws;                       // N*128
    float* xr     = xl + N * C_IN;            // N*128
    float* escore = xr + N * C_IN;            // Etot*4 (scores, then weights)
    float* mmax   = escore + Etot * 4;        // N*4
    float* denom  = mmax + N * 4;             // N*4

    {
        long total = N * 32;
        init_kernel<<<(int)((total + 255) / 256), 256, 0, stream>>>(out, bias, mmax, denom, N);
    }
    {
        int blocks = (int)((N + 15) / 16);
        ln_gemm_kernel<<<blocks, 256, 0, stream>>>(x, gamma, beta, Wl, bl, Wr, br, xl, xr, N);
    }
    {
        int blocks = (int)((Etot + 7) / 8);
        edge_score_kernel<<<blocks, 256, 0, stream>>>(srcI, dstI, xl, xr, att, escore, mmax, E, Etot);
    }
    {
        long total = Etot * 4;
        softmax_denom_kernel<<<(int)((total + 255) / 256), 256, 0, stream>>>(dstI, escore, mmax, denom, E, Etot);
    }
    {
        int blocks = (int)((Etot + 7) / 8);
        aggregate_kernel<<<blocks, 256, 0, stream>>>(srcI, dstI, xl, escore, denom, out, E, Etot);
    }
}